// LossFunc_69372311765146
// MI455X (gfx1250) — compile-verified
//
#include <hip/hip_runtime.h>
#include <hip/hip_bf16.h>

#define TAU_F 0.1f
#define EPS_F 1e-12f

typedef __attribute__((ext_vector_type(2))) float v2f;
typedef __attribute__((ext_vector_type(8))) float v8f;

struct RouteCfg { float w; int idx; int i0; int i1; };

// ---------------------------------------------------------------------------
// Kernel 1: tiny routing config (only state NUM_STATES-1 matters — the loop in
// the reference overwrites `last` every iteration).
// ---------------------------------------------------------------------------
__global__ void route_cfg_kernel(const float* __restrict__ a_ops,
                                 const float* __restrict__ a_opr,
                                 const float* __restrict__ g_ops,
                                 const float* __restrict__ g_opr,
                                 RouteCfg* __restrict__ cfg, int state) {
    const int base = state * 8;
    float s[8], t[8];
    for (int j = 0; j < 8; ++j) {
        s[j] = (a_ops[base + j] + g_ops[base + j]) * (1.0f / TAU_F);
        t[j] = (a_opr[base + j] + g_opr[base + j]) * (1.0f / TAU_F);
    }
    // argmax of ops scores (softmax is monotone; ties -> first index)
    int idx = 0; float sm = s[0];
    for (int j = 1; j < 8; ++j) if (s[j] > sm) { sm = s[j]; idx = j; }
    float den = 0.f;
    for (int j = 0; j < 8; ++j) den += __expf(s[j] - sm);
    float w = 1.0f / den;                    // softmax value at the argmax
    // top-2 of operator scores, stable tie-break like jax.lax.top_k
    int i0 = 0; float t0 = t[0];
    for (int j = 1; j < 8; ++j) if (t[j] > t0) { t0 = t[j]; i0 = j; }
    int i1 = (i0 == 0) ? 1 : 0; float t1 = t[i1];
    for (int j = 0; j < 8; ++j) {
        if (j == i0) continue;
        if (t[j] > t1) { t1 = t[j]; i1 = j; }
    }
    cfg->w = w; cfg->idx = idx; cfg->i0 = i0; cfg->i1 = i1;
}

// ---------------------------------------------------------------------------
// Kernel 2: one wave32 per row. Coalesced float4 loads, single-pass online
// softmax (branchy rescale: ~1 exp/elem), 4 independent streams for ILP,
// tracks m2 = max{ l_j : l_j < l_target }. Deterministic block partials.
// ---------------------------------------------------------------------------
__global__ void __launch_bounds__(256)
row_loss_kernel(const float* __restrict__ logits,
                const int* __restrict__ target,
                const RouteCfg* __restrict__ cfg,
                float* __restrict__ partials,
                int n, int C) {
    const int lane  = threadIdx.x & 31;
    const int wid   = threadIdx.x >> 5;
    const int wpb   = blockDim.x >> 5;
    const int gwave = blockIdx.x * wpb + wid;
    const int nwave = gridDim.x * wpb;

    const float w  = cfg->w;
    const int  idx = cfg->idx;
    const int  i0  = cfg->i0;
    const int  i1  = cfg->i1;

    const int C4 = C >> 2;
    float acc = 0.0f;

    for (int row = gwave; row < n; row += nwave) {
        const float* rp = logits + (size_t)row * (size_t)C;
        const float  lk = rp[target[row]];          // uniform address per wave

        // 4 independent online-softmax streams (break the serial exp chain)
        float M0 = -INFINITY, M1 = -INFINITY, M2s = -INFINITY, M3 = -INFINITY;
        float S0 = 0.f, S1 = 0.f, S2 = 0.f, S3 = 0.f;
        float Mj = -INFINITY;                        // max of logits strictly below lk

        auto proc = [&](float v, float& M, float& S) {
            if (v > M) { S = S * __expf(M - v) + 1.0f; M = v; }   // rare path
            else       { S += __expf(v - M); }                    // common path
            Mj = fmaxf(Mj, (v < lk) ? v : -INFINITY);
        };

        const float4* rv = (const float4*)rp;        // rows are 16B-aligned (C=1000)
        for (int j = lane; j < C4; j += 32) {
            float4 v = rv[j];
            proc(v.x, M0, S0);
            proc(v.y, M1, S1);
            proc(v.z, M2s, S2);
            proc(v.w, M3, S3);
        }
        for (int j = (C4 << 2) + lane; j < C; j += 32)   // generic tail
            proc(rp[j], M0, S0);

        auto comb = [](float& M, float& S, float oM, float oS) {
            float nm = fmaxf(M, oM);
            S = S * __expf(M - nm) + oS * __expf(oM - nm);
            M = nm;
        };
        comb(M0, S0, M1, S1);
        comb(M2s, S2, M3, S3);
        comb(M0, S0, M2s, S2);

        // wave32 butterfly merge
        for (int off = 16; off; off >>= 1) {
            float oM = __shfl_xor(M0, off, 32);
            float oS = __shfl_xor(S0, off, 32);
            float oJ = __shfl_xor(Mj, off, 32);
            comb(M0, S0, oM, oS);
            Mj = fmaxf(Mj, oJ);
        }

        const float logZ  = M0 + __logf(S0);
        const float logpk = lk - logZ;
        const float pk    = __expf(logpk);
        const float pj    = (Mj == -INFINITY) ? 0.0f : __expf(Mj - logZ);

        // choices = [pk, pj, 1, pk, pj, 1, pk, pj]  -> period 3
        const float ch[3] = { pk, pj, 1.0f };
        const float x1 = ch[i0 % 3];
        const float x2 = ch[i1 % 3];
        float res;
        switch (idx) {
            case 0: res = x1 + x2; break;
            case 1: res = x1 * x2; break;
            case 2: res = x1 - x2; break;
            case 3: res = x1 / (x2 + EPS_F); break;
            case 4: res = fmaxf(x1, x2); break;
            case 5: res = fminf(x1, x2); break;
            case 6: res = x1 / (1.0f + __expf(-x2)); break;   // x1 * sigmoid(x2)
            default: res = fabsf(x1 - x2); break;
        }
        const float last = w * res;
        const float l2 = last * last;
        const float l4 = l2 * l2;
        acc += -(l4 * last) * logpk;                 // GAMMA = 5
    }

    __shared__ float sacc[8];
    if (lane == 0) sacc[wid] = acc;
    __syncthreads();
    if (threadIdx.x == 0) {
        float s = 0.f;
        for (int i = 0; i < wpb; ++i) s += sacc[i];  // fixed order: deterministic
        partials[blockIdx.x] = s;
    }
}

// ---------------------------------------------------------------------------
// Kernel 3: deterministic final reduction, finished with V_WMMA_F32_16X16X4_F32
// (A = 16x4 f32 holding 64 lane-accumulators, B = ones -> D[m,n] = rowsum(m)).
// ---------------------------------------------------------------------------
__global__ void final_reduce_kernel(const float* __restrict__ partials, int P,
                                    float* __restrict__ out) {
    const int lane = threadIdx.x;                    // launched with 32 threads
    float a0 = 0.f, a1 = 0.f;
    for (int base = 0; base < P; base += 64) {       // P is a multiple of 64
        a0 += partials[base + lane];
        a1 += partials[base + 32 + lane];
    }
#if __has_builtin(__builtin_amdgcn_wmma_f32_16x16x4_f32)
    v2f A; A[0] = a0; A[1] = a1;                     // 16x4 f32 A-matrix
    v2f B; B[0] = 1.0f; B[1] = 1.0f;                 // all-ones 4x16 B-matrix
    v8f Cm = {};
    v8f D = __builtin_amdgcn_wmma_f32_16x16x4_f32(
        /*neg_a=*/false, A, /*neg_b=*/false, B,
        /*c_mod=*/(short)0, Cm, /*reuse_a=*/false, /*reuse_b=*/false);
    // D[m,n] = rowsum(m) for all n.  lane0 holds D[0..7, 0], lane16 D[8..15, 0].
    float t = D[0] + D[1] + D[2] + D[3] + D[4] + D[5] + D[6] + D[7];
    float tot = __shfl(t, 0, 32) + __shfl(t, 16, 32);
#else
    float t = a0 + a1;
    for (int off = 16; off; off >>= 1) t += __shfl_xor(t, off, 32);
    float tot = t;
#endif
    if (lane == 0) out[0] = tot;
}

// ---------------------------------------------------------------------------
extern "C" void kernel_launch(void* const* d_in, const int* in_sizes, int n_in,
                              void* d_out, int out_size, void* d_ws, size_t ws_size,
                              hipStream_t stream) {
    const float* logits = (const float*)d_in[0];
    const int*   target = (const int*)d_in[1];
    const float* a_ops  = (const float*)d_in[2];
    const float* a_opr  = (const float*)d_in[3];
    const float* g_ops  = (const float*)d_in[4];
    const float* g_opr  = (const float*)d_in[5];

    const int n = in_sizes[1];                 // 262144 rows
    const int C = in_sizes[0] / n;             // 1000 classes
    const int numStates = in_sizes[2] / 8;     // 11

    RouteCfg* cfg      = (RouteCfg*)d_ws;
    float*    partials = (float*)((char*)d_ws + 256);

    const int BLOCKS = 2048;                   // 16384 waves; 16 rows/wave

    route_cfg_kernel<<<1, 1, 0, stream>>>(a_ops, a_opr, g_ops, g_opr, cfg,
                                          numStates - 1);
    row_loss_kernel<<<BLOCKS, 256, 0, stream>>>(logits, target, cfg, partials,
                                                n, C);
    final_reduce_kernel<<<1, 32, 0, stream>>>(partials, BLOCKS, (float*)d_out);
}